// EMMaskGenerator_22007412424947
// MI455X (gfx1250) — compile-verified
//
#include <hip/hip_runtime.h>
#include <math.h>

typedef __attribute__((ext_vector_type(2))) float v2f;
typedef __attribute__((ext_vector_type(8))) float v8f;

#define EM_B   8
#define EM_K   5
#define EM_HW  65536
#define EM_EPS 1.1920929e-07f
#define BLKB   16      // moment-reduction blocks per batch
#define WAVES  8       // waves per block (blockDim = 256, wave32)
#define EM_STEPS 5

// ---------------------------------------------------------------- zero fill
__global__ __launch_bounds__(256) void em_zero(float* __restrict__ p, int n) {
  int i = blockIdx.x * 256 + threadIdx.x;
  if (i < n) p[i] = 0.f;
}

// ---------------------------------------------------------------- moments
// partials layout: [B][BLKB][K][49]   (49 = 7x7 moment matrix, row-major)
__global__ __launch_bounds__(256) void em_moments(
    const float* __restrict__ images,
    const float* __restrict__ shifted,
    const float* __restrict__ pred,
    const float* __restrict__ gen,
    float* __restrict__ partials)
{
  const int b    = blockIdx.x;
  const int blk  = blockIdx.y;
  const int tid  = threadIdx.x;
  const int wave = tid >> 5;
  const int lane = tid & 31;

  // staging: per-wave 64 pixels x 12 floats (v[7], w[5]); reused for reduction
  __shared__ float smem[WAVES * 64 * 12];
  float* wlds = &smem[wave * 64 * 12];

  const float* imgB = images  + (size_t)b * 3 * EM_HW;
  const float* shfB = shifted + (size_t)b * 3 * EM_HW;
  const float* prdB = pred    + (size_t)b * EM_K * EM_HW;
  const float* genB = gen     + (size_t)b * EM_K * EM_HW;

  v8f acc0 = {}; v8f acc1 = {}; v8f acc2 = {};   // k-groups {0,1},{2,3},{4,-}

  const int pixPerWave = EM_HW / (BLKB * WAVES);     // 512
  const int base = (blk * WAVES + wave) * pixPerWave;

  // --- loop-invariant lane roles (all LDS indices pre-clamped in-bounds,
  //     validity applied as 0/1 float multipliers -> no divergent branches)
  const int   n    = lane & 15;                 // A row / B column
  const int   hi   = (lane >= 16);              // half-wave -> pixel pair
  const int   idxN = (n < 7) ? n : 0;           // B component index (clamped)
  const float mB   = (n < 7) ? 1.f : 0.f;       // B valid mask
  const int   g    = (n < 7) ? 0 : 1;           // k-group slot in each WMMA
  const int   iV   = (n < 14) ? ((n < 7) ? n : n - 7) : 0;  // A component idx
  const float m01  = (n < 14) ? 1.f : 0.f;                  // rows 14,15 unused
  const float m2   = (n < 14 && g == 0) ? 1.f : 0.f;        // k=5 slot unused
  const int   kw0  = 7 + ((n < 14) ? g     : 0);            // weight idx, c=0
  const int   kw1  = 7 + ((n < 14) ? 2 + g : 0);            // weight idx, c=1
  const int   kw2  = 7 + ((n < 14 && g == 0) ? 4 : 0);      // weight idx, c=2
  const float* laneA0 = &wlds[(hi ? 2 : 0) * 12];  // pixel for .x (K=0 / K=2)
  const float* laneB0 = &wlds[(hi ? 3 : 1) * 12];  // pixel for .y (K=1 / K=3)

  for (int it = 0; it < pixPerWave / 64; ++it) {
    const int cbase = base + it * 64;
    // ---- phase 1: stage 64 pixels (2 per lane), softmax over K=5
    #pragma unroll
    for (int h = 0; h < 2; ++h) {
      const int q = cbase + h * 32 + lane;
      float x0 = prdB[0*EM_HW+q] + genB[0*EM_HW+q];
      float x1 = prdB[1*EM_HW+q] + genB[1*EM_HW+q];
      float x2 = prdB[2*EM_HW+q] + genB[2*EM_HW+q];
      float x3 = prdB[3*EM_HW+q] + genB[3*EM_HW+q];
      float x4 = prdB[4*EM_HW+q] + genB[4*EM_HW+q];
      float mx = fmaxf(fmaxf(fmaxf(x0,x1), fmaxf(x2,x3)), x4);
      float e0 = __expf(x0-mx), e1 = __expf(x1-mx), e2 = __expf(x2-mx),
            e3 = __expf(x3-mx), e4 = __expf(x4-mx);
      float inv = 1.f / (e0+e1+e2+e3+e4);
      float* p = &wlds[(h * 32 + lane) * 12];
      p[0] = shfB[0*EM_HW+q];  p[1] = shfB[1*EM_HW+q];  p[2] = shfB[2*EM_HW+q];
      p[3] = imgB[0*EM_HW+q];  p[4] = imgB[1*EM_HW+q];  p[5] = imgB[2*EM_HW+q];
      p[6] = 1.f;
      p[7]  = e0*inv + EM_EPS; p[8]  = e1*inv + EM_EPS; p[9] = e2*inv + EM_EPS;
      p[10] = e3*inv + EM_EPS; p[11] = e4*inv + EM_EPS;
    }
    __syncthreads();
    // ---- phase 2: 16 chunks of 4 pixels, 3 WMMAs each (branch-free)
    //   D[m][n] += sum_pix  w_k(pix)*v_{m%7}(pix) * v_n(pix)
    for (int sub = 0; sub < 16; ++sub) {
      const float* eA = laneA0 + sub * 48;   // 4 pixels * 12 floats
      const float* eB = laneB0 + sub * 48;
      v2f bm;
      bm.x = eA[idxN] * mB;
      bm.y = eB[idxN] * mB;
      float vA  = eA[iV],      vB  = eB[iV];
      float vAm = vA * m01,    vBm = vB * m01;     // masked component, c=0/1
      float vA2 = vA * m2,     vB2 = vB * m2;      // masked component, c=2
      float wA0 = eA[kw0], wB0 = eB[kw0];
      float wA1 = eA[kw1], wB1 = eB[kw1];
      float wA2 = eA[kw2], wB2 = eB[kw2];
      v2f am;
      am.x = wA0 * vAm;  am.y = wB0 * vBm;
      acc0 = __builtin_amdgcn_wmma_f32_16x16x4_f32(false, am, false, bm,
                                                   (short)0, acc0, false, false);
      am.x = wA1 * vAm;  am.y = wB1 * vBm;
      acc1 = __builtin_amdgcn_wmma_f32_16x16x4_f32(false, am, false, bm,
                                                   (short)0, acc1, false, false);
      am.x = wA2 * vA2;  am.y = wB2 * vB2;
      acc2 = __builtin_amdgcn_wmma_f32_16x16x4_f32(false, am, false, bm,
                                                   (short)0, acc2, false, false);
    }
    __syncthreads();
  }

  // ---- phase 3: deterministic in-block reduction (no atomics)
  // dump accumulators: smem[wave][c][lane*8 + r]
  {
    float* wr = &smem[wave * 768];
    #pragma unroll
    for (int r = 0; r < 8; ++r) {
      wr[0*256 + lane*8 + r] = acc0[r];
      wr[1*256 + lane*8 + r] = acc1[r];
      wr[2*256 + lane*8 + r] = acc2[r];
    }
  }
  __syncthreads();
  if (tid < 294) {                 // 3 accs * 14 rows * 7 cols
    int c = tid / 98, rem = tid % 98;
    int m = rem / 7, nn = rem % 7;
    int k = 2 * c + ((m >= 7) ? 1 : 0);
    if (k < EM_K) {
      // C/D layout: value (M=m,N=nn) lives in lane nn+(m>=8?16:0), VGPR m&7
      int lanei = nn + ((m >= 8) ? 16 : 0);
      int r = m & 7;
      float s = 0.f;
      for (int w = 0; w < WAVES; ++w)
        s += smem[w * 768 + c * 256 + lanei * 8 + r];
      partials[((size_t)(b * BLKB + blk) * EM_K + k) * 49 + (m % 7) * 7 + nn] = s;
    }
  }
}

// ---------------------------------------------------------------- reduce partials
__global__ __launch_bounds__(256) void em_reduce(const float* __restrict__ partials,
                                                 float* __restrict__ moments) {
  int b = blockIdx.x;
  int t = threadIdx.x;           // 0..244  (t = k*49 + ij)
  if (t < 245) {
    float s = 0.f;
    for (int blk = 0; blk < BLKB; ++blk)
      s += partials[(size_t)(b * BLKB + blk) * 245 + t];
    moments[(size_t)b * 245 + t] = s;
  }
}

// ---------------------------------------------------------------- per-(b,k) solve
// params layout: [B][K][22] = O(12) row-major 3x4, siginv(9) 3x3, logdet(1)
__global__ __launch_bounds__(64) void em_solve(const float* __restrict__ moments,
                                               float* __restrict__ params) {
  int t = threadIdx.x;
  if (t >= EM_B * EM_K) return;
  const float* M = moments + (size_t)t * 49;

  float norm = M[3*7+3] + M[4*7+4] + M[5*7+5] + M[6*7+6];
  float invn = 1.f / norm;
  float Fn[3][4], Sn[4][4], Inv[4][4];
  for (int a = 0; a < 3; ++a)
    for (int c = 0; c < 4; ++c) Fn[a][c] = M[a*7 + 3 + c] * invn;
  for (int c = 0; c < 4; ++c)
    for (int d = 0; d < 4; ++d) {
      Sn[c][d]  = M[(3+c)*7 + 3 + d] * invn + ((c == d) ? EM_EPS : 0.f);
      Inv[c][d] = (c == d) ? 1.f : 0.f;
    }
  // Gauss-Jordan (SPD, no pivoting)
  for (int col = 0; col < 4; ++col) {
    float ip = 1.f / Sn[col][col];
    for (int j = 0; j < 4; ++j) { Sn[col][j] *= ip; Inv[col][j] *= ip; }
    for (int r = 0; r < 4; ++r) if (r != col) {
      float f = Sn[r][col];
      for (int j = 0; j < 4; ++j) { Sn[r][j] -= f * Sn[col][j]; Inv[r][j] -= f * Inv[col][j]; }
    }
  }
  float O[3][4];
  for (int a = 0; a < 3; ++a)
    for (int d = 0; d < 4; ++d) {
      float s = 0.f;
      for (int c = 0; c < 4; ++c) s += Fn[a][c] * Inv[c][d];
      O[a][d] = s;
    }
  // sigma = (Mss - O*Msp^T - (O*Msp^T)^T + O*Mpp*O^T)/Z + eps*I3
  float Z = M[6*7+6], izZ = 1.f / Z;
  float T[3][3], U[3][4], Sg[3][3];
  for (int a = 0; a < 3; ++a)
    for (int bb = 0; bb < 3; ++bb) {
      float s = 0.f;
      for (int c = 0; c < 4; ++c) s += O[a][c] * M[bb*7 + 3 + c];
      T[a][bb] = s;
    }
  for (int a = 0; a < 3; ++a)
    for (int d = 0; d < 4; ++d) {
      float s = 0.f;
      for (int c = 0; c < 4; ++c) s += O[a][c] * M[(3+c)*7 + 3 + d];
      U[a][d] = s;
    }
  for (int a = 0; a < 3; ++a)
    for (int bb = 0; bb < 3; ++bb) {
      float q = 0.f;
      for (int d = 0; d < 4; ++d) q += U[a][d] * O[bb][d];
      Sg[a][bb] = (M[a*7 + bb] - T[a][bb] - T[bb][a] + q) * izZ
                + ((a == bb) ? EM_EPS : 0.f);
    }
  float c00 = Sg[1][1]*Sg[2][2] - Sg[1][2]*Sg[2][1];
  float c01 = Sg[1][0]*Sg[2][2] - Sg[1][2]*Sg[2][0];
  float c02 = Sg[1][0]*Sg[2][1] - Sg[1][1]*Sg[2][0];
  float det = Sg[0][0]*c00 - Sg[0][1]*c01 + Sg[0][2]*c02;
  float id  = 1.f / det;
  float SI[3][3];
  SI[0][0] =  c00*id;
  SI[0][1] = -(Sg[0][1]*Sg[2][2] - Sg[0][2]*Sg[2][1])*id;
  SI[0][2] =  (Sg[0][1]*Sg[1][2] - Sg[0][2]*Sg[1][1])*id;
  SI[1][0] = -c01*id;
  SI[1][1] =  (Sg[0][0]*Sg[2][2] - Sg[0][2]*Sg[2][0])*id;
  SI[1][2] = -(Sg[0][0]*Sg[1][2] - Sg[0][2]*Sg[1][0])*id;
  SI[2][0] =  c02*id;
  SI[2][1] = -(Sg[0][0]*Sg[2][1] - Sg[0][1]*Sg[2][0])*id;
  SI[2][2] =  (Sg[0][0]*Sg[1][1] - Sg[0][1]*Sg[1][0])*id;

  float* P = params + (size_t)t * 22;
  for (int a = 0; a < 3; ++a)
    for (int d = 0; d < 4; ++d) P[a*4 + d] = O[a][d];
  for (int a = 0; a < 3; ++a)
    for (int bb = 0; bb < 3; ++bb) P[12 + a*3 + bb] = SI[a][bb];
  P[21] = logf(det);
}

// ---------------------------------------------------------------- E-step
__global__ __launch_bounds__(256) void em_estep(
    const float* __restrict__ images,
    const float* __restrict__ shifted,
    const float* __restrict__ params,
    float* __restrict__ gen)
{
  int idx = blockIdx.x * 256 + threadIdx.x;   // over B*HW pixels
  int b = idx / EM_HW;
  int q = idx - b * EM_HW;
  const float* imgB = images  + (size_t)b * 3 * EM_HW;
  const float* shfB = shifted + (size_t)b * 3 * EM_HW;
  float i0 = imgB[0*EM_HW+q], i1 = imgB[1*EM_HW+q], i2 = imgB[2*EM_HW+q];
  float s0 = shfB[0*EM_HW+q], s1 = shfB[1*EM_HW+q], s2 = shfB[2*EM_HW+q];
  #pragma unroll
  for (int k = 0; k < EM_K; ++k) {
    const float* P = params + (size_t)(b * EM_K + k) * 22;
    float d0 = s0 - (P[0]*i0 + P[1]*i1 + P[2]*i2  + P[3]);
    float d1 = s1 - (P[4]*i0 + P[5]*i1 + P[6]*i2  + P[7]);
    float d2 = s2 - (P[8]*i0 + P[9]*i1 + P[10]*i2 + P[11]);
    float m0 = P[12]*d0 + P[13]*d1 + P[14]*d2;
    float m1 = P[15]*d0 + P[16]*d1 + P[17]*d2;
    float m2 = P[18]*d0 + P[19]*d1 + P[20]*d2;
    float maha = d0*m0 + d1*m1 + d2*m2;
    gen[((size_t)b * EM_K + k) * EM_HW + q] = -0.5f * maha - 0.5f * P[21];
  }
}

// ---------------------------------------------------------------- driver
extern "C" void kernel_launch(void* const* d_in, const int* in_sizes, int n_in,
                              void* d_out, int out_size, void* d_ws, size_t ws_size,
                              hipStream_t stream) {
  const float* images  = (const float*)d_in[0];
  const float* shifted = (const float*)d_in[1];
  const float* pred    = (const float*)d_in[2];
  float* gen = (float*)d_out;

  float* partials = (float*)d_ws;                         // B*BLKB*245
  float* moments  = partials + (size_t)EM_B * BLKB * 245; // B*245
  float* params   = moments  + (size_t)EM_B * 245;        // B*K*22

  const int nGen = EM_B * EM_K * EM_HW;
  em_zero<<<(nGen + 255) / 256, 256, 0, stream>>>(gen, nGen);

  for (int step = 0; step < EM_STEPS; ++step) {
    em_moments<<<dim3(EM_B, BLKB), 256, 0, stream>>>(images, shifted, pred, gen, partials);
    em_reduce<<<EM_B, 256, 0, stream>>>(partials, moments);
    em_solve<<<1, 64, 0, stream>>>(moments, params);
    em_estep<<<(EM_B * EM_HW) / 256, 256, 0, stream>>>(images, shifted, params, gen);
  }
}